// MLPKANlayer_22789096472783
// MI455X (gfx1250) — compile-verified
//
#include <hip/hip_runtime.h>

typedef float v8f __attribute__((ext_vector_type(8)));
typedef float v2f __attribute__((ext_vector_type(2)));

#define IN_SIZE  128
#define OUT_SIZE 128
#define HW       5      // subnet hidden width
#define BATCH    2048
#define BT       16     // batch tile = WMMA M
#define OW       8      // output features per block (one per wave)

// Fast SiLU: v * 1/(1 + exp2(-v*log2e)) -> v_mul + v_exp_f32 + v_add + v_rcp_f32 + v_mul
__device__ __forceinline__ float silu_fast(float v) {
    float e = __builtin_amdgcn_exp2f(v * -1.44269504088896f);
    return v * __builtin_amdgcn_rcpf(1.0f + e);
}

__global__ __launch_bounds__(256)
void kan_fused_wmma(const float* __restrict__ x,
                    const float* __restrict__ W0, const float* __restrict__ b0,
                    const float* __restrict__ W1, const float* __restrict__ b1,
                    const float* __restrict__ W2, const float* __restrict__ b2,
                    float* __restrict__ out)
{
    __shared__ float xs[BT][IN_SIZE + 1];   // +1 pad: conflict-free column reads

    const int bt   = blockIdx.x;            // batch tile index
    const int og   = blockIdx.y;            // group of 8 output features
    const int tid  = threadIdx.x;
    const int wave = tid >> 5;
    const int lane = tid & 31;
    const int lane_lo = lane & 15;          // WMMA row (batch) / col (j) index
    const int hi      = lane >> 4;          // 0: K=0/1 half, 1: K=2/3 half

    const int brow0 = bt * BT;
    const int o     = og * OW + wave;

    // Stage x[brow0 : brow0+16, 0:128] into LDS, fully coalesced.
    for (int t = tid; t < BT * IN_SIZE; t += 256) {
        const int r = t >> 7;               // /128
        const int c = t & (IN_SIZE - 1);
        xs[r][c] = x[(brow0 + r) * IN_SIZE + c];
    }
    __syncthreads();

    const int  j   = lane_lo;               // hidden-unit column in B/D
    const int  jc  = (j < HW) ? j : 0;      // clamped index -> loads never OOB
    const float mj  = (j < HW)         ? 1.0f : 0.0f;  // column-valid mask
    const float m4  = (j < HW && !hi)  ? 1.0f : 0.0f;  // K=4 B-row mask (lo half only)
    const float mlo = hi               ? 0.0f : 1.0f;  // lo-half mask for A K=4

    const int ka0 = hi ? 2 : 0;             // K carried in A/B vgpr0 for this half
    const int ka1 = hi ? 3 : 1;             // K carried in A/B vgpr1 for this half

    float acc[8];
    #pragma unroll
    for (int r = 0; r < 8; ++r) acc[r] = 0.0f;
    float bsum = 0.0f;                      // sum_i b2[i*128+o]

    const v8f zeroc = {};                   // loop-invariant zero C operand

    for (int i = 0; i < IN_SIZE; ++i) {
        const int n = i * OUT_SIZE + o;     // subnet id (wave-uniform)

        const float* w0n = W0 + n * HW;
        const float* b0n = b0 + n * HW;
        const float* w1n = W1 + n * HW * HW;

        // Branch-free prefetch of next iteration's W1 (clamped to stay in-array).
        const int npf = n + ((i + 1 < IN_SIZE) ? OUT_SIZE : 0);
        __builtin_prefetch(W1 + npf * HW * HW, 0, 0);

        // Per-lane batch scalar u (lane & lane+16 share the same batch row).
        const float u = xs[lane_lo][i];

        // ---- layer 0: h0[k] = silu(u*W0[k] + b0[k]), laid out as WMMA A ----
        const float ha0 = silu_fast(u * w0n[ka0] + b0n[ka0]);
        const float ha1 = silu_fast(u * w0n[ka1] + b0n[ka1]);
        const float hk4 = mlo * silu_fast(u * w0n[4] + b0n[4]);  // K=4 slice

        // ---- B = W1^T : B[k][j] = W1[j][k]; mask via mul, loads unconditional
        const float bw0 = mj * w1n[jc * HW + ka0];
        const float bw1 = mj * w1n[jc * HW + ka1];
        const float bw4 = m4 * w1n[jc * HW + 4];

        const float cb  = mj * b1[n * HW + jc];   // b1[j], folded in post-WMMA
        const float w2l = mj * W2[n * HW + jc];   // W2[j]

        // ---- layer 1: h1pre^T (16x16) = h0^T (16x5) @ W1^T (5x16) ----------
        v2f a1v; a1v.x = ha0; a1v.y = ha1;
        v2f b1v; b1v.x = bw0; b1v.y = bw1;
        v8f d = __builtin_amdgcn_wmma_f32_16x16x4_f32(
                    false, a1v, false, b1v, (short)0, zeroc, false, false);

        v2f a2v; a2v.x = hk4; a2v.y = 0.0f;   // K=4 slice; K=5..7 padded to 0
        v2f b2v; b2v.x = bw4; b2v.y = 0.0f;
        d = __builtin_amdgcn_wmma_f32_16x16x4_f32(
                    false, a2v, false, b2v, (short)0, d, false, false);

        // ---- layer 2 partial: this lane's column-j contribution, all 8 rows.
        // Cross-lane j-reduction is deferred to after the i-loop (sums commute).
        #pragma unroll
        for (int r = 0; r < 8; ++r) {
            acc[r] += w2l * silu_fast(d[r] + cb);   // invalid cols: exact 0
        }

        bsum += b2[n];
    }

    // One-time reduction over the 16 j-lanes of each half (rows r / 8+r).
    #pragma unroll
    for (int r = 0; r < 8; ++r) {
        float t = acc[r];
        t += __shfl_xor(t, 1, 32);
        t += __shfl_xor(t, 2, 32);
        t += __shfl_xor(t, 4, 32);
        t += __shfl_xor(t, 8, 32);
        acc[r] = t;
    }

    // Lanes 0 (rows 0..7) and 16 (rows 8..15) publish this wave's column o.
    if (lane_lo == 0) {
        #pragma unroll
        for (int r = 0; r < 8; ++r) {
            out[(brow0 + hi * 8 + r) * OUT_SIZE + o] = acc[r] + bsum;
        }
    }
}

extern "C" void kernel_launch(void* const* d_in, const int* in_sizes, int n_in,
                              void* d_out, int out_size, void* d_ws, size_t ws_size,
                              hipStream_t stream) {
    const float* x  = (const float*)d_in[0];
    const float* W0 = (const float*)d_in[1];
    const float* b0 = (const float*)d_in[2];
    const float* W1 = (const float*)d_in[3];
    const float* b1 = (const float*)d_in[4];
    const float* W2 = (const float*)d_in[5];
    const float* b2 = (const float*)d_in[6];
    float* out = (float*)d_out;

    dim3 grid(BATCH / BT, OUT_SIZE / OW);   // 128 batch tiles x 16 o-groups
    kan_fused_wmma<<<grid, dim3(256), 0, stream>>>(x, W0, b0, W1, b1, W2, b2, out);
}